// LogNeuralJastrowSlater_62268435857711
// MI455X (gfx1250) — compile-verified
//
#include <hip/hip_runtime.h>
#include <math.h>

#define SIZE 128
#define NF 64
#define HIDDEN 128

typedef __attribute__((ext_vector_type(2))) float v2f;
typedef __attribute__((ext_vector_type(8))) float v8f;

__device__ __forceinline__ v8f wmma_f32_16x16x4(v2f a, v2f b, v8f c) {
  // D = A(16x4) * B(4x16) + C(16x16), fp32. 8-arg form:
  // (neg_a, A, neg_b, B, c_mod, C, reuse_a, reuse_b) — all modifiers 0.
  return __builtin_amdgcn_wmma_f32_16x16x4_f32(false, a, false, b, (short)0, c,
                                               false, false);
}

// ---------------------------------------------------------------------------
// Kernel 1: J[b] = sum_h tanh((n @ W)[b,h] + bias[h]) for a tile of 16 samples
// per wave. Writes J into the REAL slot of the complex64 output; kernel 2 adds
// the log-determinant on top. One wave (32 threads) per block.
// ---------------------------------------------------------------------------
__global__ __launch_bounds__(32) void jastrow_kernel(
    const int* __restrict__ n, const float* __restrict__ W,
    const float* __restrict__ bias, float* __restrict__ out) {
  __shared__ float As[16][SIZE + 1];  // 16 samples x 128 occupations (padded)
  __shared__ float Ws[SIZE][17];      // 128 x 16 slab of W (padded)

  const int lane = threadIdx.x;
  const int tile = blockIdx.x;
  const int m = lane & 15;   // N-lane within 16
  const int h = lane >> 4;   // half select

  const int* nrow = n + (size_t)tile * 16 * SIZE;
  for (int t = lane; t < 16 * SIZE; t += 32)
    As[t >> 7][t & 127] = (float)nrow[t];

  float jsum[8];
#pragma unroll
  for (int v = 0; v < 8; ++v) jsum[v] = 0.f;

  for (int jt = 0; jt < HIDDEN / 16; ++jt) {
    __syncthreads();
    for (int t = lane; t < SIZE * 16; t += 32)
      Ws[t >> 4][t & 15] = W[(t >> 4) * HIDDEN + jt * 16 + (t & 15)];
    __syncthreads();

    v8f c = {0.f, 0.f, 0.f, 0.f, 0.f, 0.f, 0.f, 0.f};
#pragma unroll
    for (int kc = 0; kc < SIZE / 4; ++kc) {
      const int k = kc * 4 + 2 * h;
      v2f a, b;
      a.x = As[m][k];                 // A fragment: row m, K = 2h / 2h+1
      a.y = As[m][k + 1];
      b.x = Ws[k][m];                 // B fragment: K rows, col m
      b.y = Ws[k + 1][m];
      c = wmma_f32_16x16x4(a, b, c);
    }
    const float bj = bias[jt * 16 + m];
#pragma unroll
    for (int v = 0; v < 8; ++v) jsum[v] += tanhf(c[v] + bj);
  }

  // Reduce over the 16 N-lanes inside each half (C/D: lane holds M = v + 8h).
#pragma unroll
  for (int v = 0; v < 8; ++v) {
    float s = jsum[v];
#pragma unroll
    for (int off = 1; off < 16; off <<= 1) s += __shfl_xor(s, off, 32);
    jsum[v] = s;
  }
  if (m == 0) {
#pragma unroll
    for (int v = 0; v < 8; ++v) {
      const int samp = tile * 16 + h * 8 + v;
      out[2 * samp] = jsum[v];  // real slot <- J ; kernel 2 adds logabs(det)
    }
  }
}

// ---------------------------------------------------------------------------
// Kernel 2: per-sample slogdet of A = M[R] (64x64) via blocked LU with partial
// pivoting. One wave per sample. Panel factorization + triangular solve on the
// VALU; trailing-submatrix GEMM updates (2/3 of the flops) on fp32 WMMA.
// ---------------------------------------------------------------------------
__global__ __launch_bounds__(32) void slogdet_kernel(
    const int* __restrict__ n, const float* __restrict__ M,
    float* __restrict__ out) {
  __shared__ float A[NF][NF + 1];  // 64x64, padded: column access conflict-free
  __shared__ int Rs[NF];

  const int s = blockIdx.x;
  const int lane = threadIdx.x;
  const int* nrow = n + (size_t)s * SIZE;

  // --- Build R: ascending nonzero positions via wave prefix-scan -----------
  int vals[4], cnt = 0;
#pragma unroll
  for (int q = 0; q < 4; ++q) {
    vals[q] = nrow[lane * 4 + q];
    cnt += vals[q];
  }
  int incl = cnt;
#pragma unroll
  for (int off = 1; off < 32; off <<= 1) {
    int t = __shfl_up(incl, off, 32);
    if (lane >= off) incl += t;
  }
  int pos = incl - cnt;  // exclusive prefix
#pragma unroll
  for (int q = 0; q < 4; ++q)
    if (vals[q]) Rs[pos++] = lane * 4 + q;
  __syncthreads();

  // --- Gather A = M[R] ------------------------------------------------------
  for (int t = lane; t < NF * NF; t += 32) {
    const int r = t >> 6, c = t & 63;
    A[r][c] = M[Rs[r] * NF + c];
  }
  __syncthreads();

  // --- Blocked LU (NB = 16) ------------------------------------------------
  float logabs = 0.f;
  int neg = 0;
  const int m = lane & 15;
  const int h = lane >> 4;

  for (int k0 = 0; k0 < NF; k0 += 16) {
    // Panel factorization: columns k0 .. k0+15 (in-panel updates only).
    for (int j = k0; j < k0 + 16; ++j) {
      // Partial pivot over rows j..63 of column j (smallest index on ties).
      float best = -1.f;
      int bidx = j;
      for (int i = j + lane; i < NF; i += 32) {
        float av = fabsf(A[i][j]);
        if (av > best || (av == best && i < bidx)) { best = av; bidx = i; }
      }
#pragma unroll
      for (int off = 16; off > 0; off >>= 1) {
        float ob = __shfl_xor(best, off, 32);
        int oi = __shfl_xor(bidx, off, 32);
        if (ob > best || (ob == best && oi < bidx)) { best = ob; bidx = oi; }
      }
      const int p = bidx;  // uniform across the wave
      if (p != j) {
        for (int c = lane; c < NF; c += 32) {
          float t1 = A[j][c];
          A[j][c] = A[p][c];
          A[p][c] = t1;
        }
        neg ^= 1;
      }
      __syncthreads();

      const float d = A[j][j];  // broadcast read
      if (d < 0.f) neg ^= 1;
      logabs += logf(fabsf(d));
      const float rinv = 1.f / d;

      // Scale multipliers + rank-1 update restricted to the panel columns.
      for (int i = j + 1 + lane; i < NF; i += 32) {
        const float lij = A[i][j] * rinv;
        A[i][j] = lij;
        for (int jj = j + 1; jj < k0 + 16; ++jj)
          A[i][jj] -= lij * A[j][jj];
      }
      __syncthreads();
    }

    if (k0 < NF - 16) {
      // Triangular solve: U12 = L11^{-1} * A[k0:k0+16, k0+16:64].
      const int ncols = NF - (k0 + 16);
      for (int c0 = lane; c0 < ncols; c0 += 32) {
        const int c = k0 + 16 + c0;
        float u[16];
#pragma unroll
        for (int r = 0; r < 16; ++r) u[r] = A[k0 + r][c];
#pragma unroll
        for (int r = 1; r < 16; ++r) {
          float acc = u[r];
          for (int q = 0; q < r; ++q) acc -= A[k0 + r][k0 + q] * u[q];
          u[r] = acc;
        }
#pragma unroll
        for (int r = 0; r < 16; ++r) A[k0 + r][c] = u[r];
      }
      __syncthreads();

      // Trailing update A22 -= L21 * U12 via fp32 WMMA, K = 16 (4 chained).
      const int nt = (NF - (k0 + 16)) / 16;  // 3, 2, 1
      for (int rt = 0; rt < nt; ++rt) {
        const int rbase = k0 + 16 + rt * 16;
        for (int ct = 0; ct < nt; ++ct) {
          const int cbase = k0 + 16 + ct * 16;
          v8f c;
#pragma unroll
          for (int v = 0; v < 8; ++v) c[v] = A[rbase + v + 8 * h][cbase + m];
#pragma unroll
          for (int kc = 0; kc < 4; ++kc) {
            const int kk = k0 + kc * 4 + 2 * h;
            v2f a, b;
            a.x = -A[rbase + m][kk];      // negate L21: D = C - L21*U12
            a.y = -A[rbase + m][kk + 1];
            b.x = A[kk][cbase + m];
            b.y = A[kk + 1][cbase + m];
            c = wmma_f32_16x16x4(a, b, c);
          }
#pragma unroll
          for (int v = 0; v < 8; ++v) A[rbase + v + 8 * h][cbase + m] = c[v];
        }
      }
      __syncthreads();
    }
  }

  // --- Epilogue: out[s] = (logabs + J) + i*(pi if sign<0 else 0) -----------
  if (lane == 0) {
    const float J = out[2 * s];  // written by jastrow_kernel earlier on stream
    out[2 * s] = logabs + J;
    out[2 * s + 1] = neg ? 3.14159265358979323846f : 0.f;
  }
}

// ---------------------------------------------------------------------------
extern "C" void kernel_launch(void* const* d_in, const int* in_sizes, int n_in,
                              void* d_out, int out_size, void* d_ws,
                              size_t ws_size, hipStream_t stream) {
  const int* n = (const int*)d_in[0];     // [B, 128] int32 occupations
  const float* M = (const float*)d_in[1]; // [128, 64]
  const float* W = (const float*)d_in[2]; // [128, 128]
  const float* b = (const float*)d_in[3]; // [128]
  float* out = (float*)d_out;             // complex64 [B] as interleaved f32

  const int B = in_sizes[0] / SIZE;

  jastrow_kernel<<<B / 16, 32, 0, stream>>>(n, W, b, out);
  slogdet_kernel<<<B, 32, 0, stream>>>(n, M, out);
  (void)d_ws; (void)ws_size; (void)n_in; (void)out_size;
}